// PhysicsGNNODE_1013612281975
// MI455X (gfx1250) — compile-verified
//
#include <hip/hip_runtime.h>
#include <hip/hip_bf16.h>

#define PN   10000
#define ZN   1500
#define ZPAD 1504
#define DN   64
#define HN   2
#define EPP  160000
#define ELOC 24000
#define ETOT 170000   // EPP + PN self loops
#define HIDN 128
#define FINN 256
#define ADJW 47       // ceil(1500/32)

typedef __attribute__((ext_vector_type(2))) float v2f;
typedef __attribute__((ext_vector_type(8))) float v8f;

__device__ __forceinline__ v8f wmma4(v2f a, v2f b, v8f c) {
  // D = A(16x4 f32) x B(4x16 f32) + C(16x16 f32)
  return __builtin_amdgcn_wmma_f32_16x16x4_f32(false, a, false, b, (short)0, c,
                                               false, false);
}

__device__ __forceinline__ void atomicMaxF(float* addr, float val) {
  int* ai = (int*)addr;
  int cur = __float_as_int(*addr);
  while (__int_as_float(cur) < val) {
    int prev = atomicCAS(ai, cur, __float_as_int(val));
    if (prev == cur) break;
    cur = prev;
  }
}

// ---------------------------------------------------------------- precompute
__global__ void pre_zero_kernel(float* dinv, float* ctx, unsigned* adjm) {
  int i = blockIdx.x * blockDim.x + threadIdx.x;
  if (i < ZN) dinv[i] = 1.0f;                 // self-loop contributes 1 to degree
  if (i < ZN * DN) ctx[i] = 0.0f;
  if (i < ZN * ADJW) adjm[i] = 0u;
}

__global__ void pre_deg_kernel(const int* __restrict__ le, float* dinv) {
  int e = blockIdx.x * blockDim.x + threadIdx.x;
  if (e >= ELOC) return;
  atomicAdd(&dinv[le[ELOC + e]], 1.0f);
}

__global__ void pre_dinv_kernel(float* dinv) {
  int i = blockIdx.x * blockDim.x + threadIdx.x;
  if (i >= ZN) return;
  dinv[i] = rsqrtf(dinv[i]);                  // deg >= 1 always (self loop)
}

__global__ void pre_xw_kernel(const float* __restrict__ emb,
                              const float* __restrict__ Wgcn, float* XW) {
  int i = blockIdx.x * blockDim.x + threadIdx.x;
  if (i >= ZN * DN) return;
  int z = i >> 6, d = i & 63;
  float s = 0.0f;
  for (int k = 0; k < DN; ++k) s += emb[z * DN + k] * Wgcn[k * DN + d];
  XW[i] = s;
}

__global__ void pre_gcn_scatter_kernel(const int* __restrict__ le,
                                       const float* __restrict__ XW,
                                       const float* __restrict__ dinv, float* ctx) {
  int tid = blockIdx.x * blockDim.x + threadIdx.x;
  if (tid >= (ELOC + ZN) * DN) return;
  int e = tid >> 6, d = tid & 63;
  int s, dd;
  if (e < ELOC) { s = le[e]; dd = le[ELOC + e]; } else { s = dd = e - ELOC; }
  atomicAdd(&ctx[dd * DN + d], XW[s * DN + d] * dinv[s] * dinv[dd]);
}

__global__ void pre_adj_kernel(const int* __restrict__ le, unsigned* adjm) {
  int e = blockIdx.x * blockDim.x + threadIdx.x;
  if (e >= ELOC + ZN) return;
  int r, c;
  if (e < ELOC) { r = le[e]; c = le[ELOC + e]; } else { r = c = e - ELOC; }
  atomicOr(&adjm[r * ADJW + (c >> 5)], 1u << (c & 31));
}

__global__ void pre_embpad_kernel(const float* __restrict__ emb, float* embp) {
  int i = blockIdx.x * blockDim.x + threadIdx.x;
  if (i >= ZPAD * DN) return;
  int z = i >> 6, d = i & 63;
  embp[i] = (z < ZN) ? emb[z * DN + d] : 0.0f;
}

__global__ void pre_esq_kernel(const float* __restrict__ emb, float* esq) {
  int z = blockIdx.x * blockDim.x + threadIdx.x;
  if (z >= ZPAD) return;
  if (z >= ZN) { esq[z] = 1e30f; return; }
  float s = 0.0f;
  for (int k = 0; k < DN; ++k) { float v = emb[z * DN + k]; s += v * v; }
  esq[z] = s;
}

__global__ void pre_basep_x0_kernel(const int* __restrict__ xz,
                                    const float* __restrict__ emb,
                                    const float* __restrict__ Wpe,
                                    const float* __restrict__ bpe,
                                    float* fin, float* X, float* out) {
  int i = blockIdx.x * blockDim.x + threadIdx.x;
  if (i >= PN * DN) return;
  int p = i >> 6, d = i & 63;
  fin[(size_t)p * FINN + 128 + d] = (float)xz[p] * Wpe[d] + bpe[d];
  float x0 = emb[xz[p] * DN + d];
  X[i] = x0;
  out[i] = x0;   // trajectory slot 0
}

// ---------------------------------------------------------------- per step
// Xp = X @ W_gat  ([P,64] @ [64,128]) via WMMA f32 16x16x4
__global__ void xp_kernel(const float* __restrict__ X, const float* __restrict__ Wg,
                          float* __restrict__ Xp) {
  int lane = threadIdx.x & 31;
  int wv = threadIdx.x >> 5;          // col tile 0..7
  int rt = blockIdx.x;                // row tile 0..624
  int m = lane & 15;
  int kk = (lane >> 4) * 2;
  const float* Arow = X + (size_t)(rt * 16 + m) * DN;
  v8f c = {};
  for (int k = 0; k < DN; k += 4) {
    v2f a, b;
    a.x = Arow[k + kk];
    a.y = Arow[k + kk + 1];
    b.x = Wg[(k + kk) * HIDN + wv * 16 + m];
    b.y = Wg[(k + kk + 1) * HIDN + wv * 16 + m];
    c = wmma4(a, b, c);
  }
  int col = wv * 16 + (lane & 15);
#pragma unroll
  for (int v = 0; v < 8; ++v) {
    int row = rt * 16 + v + 8 * (lane >> 4);
    Xp[(size_t)row * HIDN + col] = c[v];
  }
}

__global__ void attn_kernel(const float* __restrict__ Xp,
                            const float* __restrict__ att_src,
                            const float* __restrict__ att_dst,
                            float* a_s, float* a_d) {
  int i = blockIdx.x * blockDim.x + threadIdx.x;
  if (i >= PN * HN) return;
  int p = i >> 1, h = i & 1;
  const float* xp = Xp + (size_t)p * HIDN + h * DN;
  float s = 0.0f, d = 0.0f;
  for (int k = 0; k < DN; ++k) {
    s += xp[k] * att_src[h * DN + k];
    d += xp[k] * att_dst[h * DN + k];
  }
  a_s[i] = s;
  a_d[i] = d;
}

__global__ void fill_kernel(float* mbuf, float* denb, float* gacc) {
  int i = blockIdx.x * blockDim.x + threadIdx.x;
  if (i < PN * HN) { mbuf[i] = -3.0e38f; denb[i] = 0.0f; }
  if (i < PN * HIDN) gacc[i] = 0.0f;
}

__device__ __forceinline__ void edge_sd(int e, const int* epp, int& s, int& d) {
  if (e < EPP) { s = epp[e]; d = epp[EPP + e]; } else { s = d = e - EPP; }
}

__global__ void edge_max_kernel(const int* __restrict__ epp,
                                const float* __restrict__ a_s,
                                const float* __restrict__ a_d, float* mbuf) {
  int tid = blockIdx.x * blockDim.x + threadIdx.x;
  if (tid >= ETOT * HN) return;
  int e = tid >> 1, h = tid & 1;
  int s, d; edge_sd(e, epp, s, d);
  float ev = a_s[s * HN + h] + a_d[d * HN + h];
  ev = ev > 0.0f ? ev : 0.2f * ev;            // leaky_relu 0.2
  atomicMaxF(&mbuf[d * HN + h], ev);
}

__global__ void edge_exp_kernel(const int* __restrict__ epp,
                                const float* __restrict__ a_s,
                                const float* __restrict__ a_d,
                                const float* __restrict__ mbuf,
                                float* exb, float* denb) {
  int tid = blockIdx.x * blockDim.x + threadIdx.x;
  if (tid >= ETOT * HN) return;
  int e = tid >> 1, h = tid & 1;
  int s, d; edge_sd(e, epp, s, d);
  float ev = a_s[s * HN + h] + a_d[d * HN + h];
  ev = ev > 0.0f ? ev : 0.2f * ev;
  float ex = __expf(ev - mbuf[d * HN + h]);
  exb[tid] = ex;
  atomicAdd(&denb[d * HN + h], ex);
}

__global__ void edge_scatter_kernel(const int* __restrict__ epp,
                                    const float* __restrict__ exb,
                                    const float* __restrict__ denb,
                                    const float* __restrict__ Xp, float* gacc) {
  int tid = blockIdx.x * blockDim.x + threadIdx.x;
  if (tid >= ETOT * DN) return;
  int e = tid >> 6, dd = tid & 63;
  int s, d; edge_sd(e, epp, s, d);
#pragma unroll
  for (int h = 0; h < HN; ++h) {
    float alpha = exb[e * HN + h] / (denb[d * HN + h] + 1e-16f);
    atomicAdd(&gacc[(size_t)d * HIDN + h * DN + dd],
              Xp[(size_t)s * HIDN + h * DN + dd] * alpha);
  }
}

// zi = argmin_z ( ||emb_z||^2 - 2 * X . emb_z )   via WMMA score tiles
__global__ void argmin_kernel(const float* __restrict__ X,
                              const float* __restrict__ embp,
                              const float* __restrict__ esq, int* zi) {
  int lane = threadIdx.x & 31;
  int wv = blockIdx.x * (blockDim.x >> 5) + (threadIdx.x >> 5);
  if (wv >= PN / 16) return;                  // wave-uniform guard
  int m = lane & 15;
  int kk = (lane >> 4) * 2;
  const float* Arow = X + (size_t)(wv * 16 + m) * DN;
  float bestv[8];
  int bestz[8];
#pragma unroll
  for (int v = 0; v < 8; ++v) { bestv[v] = 3.4e38f; bestz[v] = 0; }
  for (int zt = 0; zt < ZPAD / 16; ++zt) {
    const float* Bcol = embp + (size_t)(zt * 16 + m) * DN;  // zone row, n = lane&15
    v8f c = {};
    for (int k = 0; k < DN; k += 4) {
      v2f a, b;
      a.x = Arow[k + kk];
      a.y = Arow[k + kk + 1];
      b.x = Bcol[k + kk];
      b.y = Bcol[k + kk + 1];
      c = wmma4(a, b, c);
    }
    int z = zt * 16 + (lane & 15);
    float es = esq[z];
#pragma unroll
    for (int v = 0; v < 8; ++v) {
      float sc = es - 2.0f * c[v];
      if (sc < bestv[v]) { bestv[v] = sc; bestz[v] = z; }
    }
  }
#pragma unroll
  for (int v = 0; v < 8; ++v) {
    float bv = bestv[v];
    int bz = bestz[v];
#pragma unroll
    for (int off = 1; off < 16; off <<= 1) {
      float ov = __shfl_xor(bv, off, 32);
      int oz = __shfl_xor(bz, off, 32);
      if (ov < bv || (ov == bv && oz < bz)) { bv = ov; bz = oz; }
    }
    if ((lane & 15) == 0) {
      int row = v + 8 * (lane >> 4);
      zi[wv * 16 + row] = bz;
    }
  }
}

__global__ void assemble_kernel(const float* __restrict__ gacc,
                                const float* __restrict__ bgat,
                                const float* __restrict__ ctx,
                                const float* __restrict__ bgcn,
                                const int* __restrict__ zi,
                                const float* __restrict__ Wt,
                                const float* __restrict__ bt,
                                const float* __restrict__ times, int ti,
                                float* fin) {
  int i = blockIdx.x * blockDim.x + threadIdx.x;
  if (i >= PN * DN) return;
  int p = i >> 6, d = i & 63;
  float t = times[ti];
  float* f = fin + (size_t)p * FINN;
  f[d]       = 0.5f * (gacc[(size_t)p * HIDN + d] + gacc[(size_t)p * HIDN + DN + d]) + bgat[d];
  f[64 + d]  = ctx[zi[p] * DN + d] + bgcn[d];
  f[192 + d] = t * Wt[d] + bt[d];
}

// v = relu(fin @ W1 + b1) @ W2 + b2   via WMMA, hidden staged through LDS
__global__ void mlp_kernel(const float* __restrict__ fin,
                           const float* __restrict__ W1, const float* __restrict__ b1,
                           const float* __restrict__ W2, const float* __restrict__ b2,
                           float* __restrict__ vout) {
  __shared__ float hs[8][16][HIDN];           // 64 KB
  int lane = threadIdx.x & 31;
  int wvl = threadIdx.x >> 5;
  int wv = blockIdx.x * 8 + wvl;
  int m = lane & 15;
  int kk = (lane >> 4) * 2;
  if (wv < PN / 16) {
    const float* Arow = fin + (size_t)(wv * 16 + m) * FINN;
    for (int nt = 0; nt < 8; ++nt) {
      v8f c = {};
      for (int k = 0; k < FINN; k += 4) {
        v2f a, b;
        a.x = Arow[k + kk];
        a.y = Arow[k + kk + 1];
        b.x = W1[(k + kk) * HIDN + nt * 16 + m];
        b.y = W1[(k + kk + 1) * HIDN + nt * 16 + m];
        c = wmma4(a, b, c);
      }
      float bb = b1[nt * 16 + (lane & 15)];
#pragma unroll
      for (int v = 0; v < 8; ++v) {
        float h = c[v] + bb;
        hs[wvl][v + 8 * (lane >> 4)][nt * 16 + (lane & 15)] = h > 0.0f ? h : 0.0f;
      }
    }
  }
  __syncthreads();
  if (wv >= PN / 16) return;
  for (int nt = 0; nt < 4; ++nt) {
    v8f c = {};
    for (int k = 0; k < HIDN; k += 4) {
      v2f a, b;
      a.x = hs[wvl][m][k + kk];
      a.y = hs[wvl][m][k + kk + 1];
      b.x = W2[(k + kk) * DN + nt * 16 + m];
      b.y = W2[(k + kk + 1) * DN + nt * 16 + m];
      c = wmma4(a, b, c);
    }
    float bb = b2[nt * 16 + (lane & 15)];
#pragma unroll
    for (int v = 0; v < 8; ++v) {
      int row = wv * 16 + v + 8 * (lane >> 4);
      vout[(size_t)row * DN + nt * 16 + (lane & 15)] = c[v] + bb;
    }
  }
}

// Per-particle sparse neighbor softmax flow + Euler update. One wave / particle.
__global__ void flow_kernel(float* __restrict__ X, const float* __restrict__ vbuf,
                            const int* __restrict__ zi, const float* __restrict__ emb,
                            const unsigned* __restrict__ adjm,
                            const float* __restrict__ times, int ti,
                            float* __restrict__ out) {
  int lane = threadIdx.x & 31;
  int i = blockIdx.x * (blockDim.x >> 5) + (threadIdx.x >> 5);
  if (i >= PN) return;
  int z0 = zi[i];
  float v0 = vbuf[(size_t)i * DN + lane];
  float v1 = vbuf[(size_t)i * DN + 32 + lane];
  float ez0 = emb[z0 * DN + lane];
  float ez1 = emb[z0 * DN + 32 + lane];
  float r = v0 * ez0 + v1 * ez1;
#pragma unroll
  for (int off = 1; off < 32; off <<= 1) r += __shfl_xor(r, off, 32);
  float bself = r;

  float mrun = -3.0e38f, den = 0.0f, ssum = 0.0f, acc0 = 0.0f, acc1 = 0.0f;
  const unsigned* row = adjm + z0 * ADJW;
  for (int w = 0; w < ADJW; ++w) {
    unsigned bits = row[w];                   // uniform across the wave
    while (bits) {
      int b = __ffs(bits) - 1;
      bits &= bits - 1;
      int z = w * 32 + b;
      float e0 = emb[z * DN + lane];
      float e1 = emb[z * DN + 32 + lane];
      float pd = v0 * e0 + v1 * e1;
      float dx0 = e0 - ez0, dx1 = e1 - ez1;
      float p2 = dx0 * dx0 + dx1 * dx1;
#pragma unroll
      for (int off = 1; off < 32; off <<= 1) {
        pd += __shfl_xor(pd, off, 32);
        p2 += __shfl_xor(p2, off, 32);
      }
      float dnrm = sqrtf(p2);
      float sn = dnrm > 1e-6f ? dnrm : 1.0f;
      float proj = (pd - bself) / sn;
      if (proj > mrun) {                      // online softmax rescale
        float sc = __expf(mrun - proj);
        den *= sc; ssum *= sc; acc0 *= sc; acc1 *= sc;
        mrun = proj;
      }
      float wexp = __expf(proj - mrun);
      float wn = wexp / sn;
      den += wexp;
      ssum += wn;
      acc0 += wn * e0;
      acc1 += wn * e1;
    }
  }
  float dt = times[ti + 1] - times[ti];
  float inv = 1.0f / den;
  float fv0 = acc0 * inv - ssum * inv * ez0;
  float fv1 = acc1 * inv - ssum * inv * ez1;
  float xn0 = X[(size_t)i * DN + lane] + dt * 0.1f * fv0;
  float xn1 = X[(size_t)i * DN + 32 + lane] + dt * 0.1f * fv1;
  X[(size_t)i * DN + lane] = xn0;
  X[(size_t)i * DN + 32 + lane] = xn1;
  size_t ob = (size_t)(ti + 1) * PN * DN + (size_t)i * DN;
  out[ob + lane] = xn0;
  out[ob + 32 + lane] = xn1;
}

// ----------------------------------------------------------------- driver
extern "C" void kernel_launch(void* const* d_in, const int* in_sizes, int n_in,
                              void* d_out, int out_size, void* d_ws, size_t ws_size,
                              hipStream_t stream) {
  (void)in_sizes; (void)n_in; (void)out_size; (void)ws_size;
  const int*   x_zones = (const int*)d_in[0];
  const int*   epp     = (const int*)d_in[1];
  const int*   le      = (const int*)d_in[2];
  const float* times   = (const float*)d_in[3];
  const float* emb     = (const float*)d_in[4];
  const float* Wpe     = (const float*)d_in[5];
  const float* bpe     = (const float*)d_in[6];
  const float* Wgat    = (const float*)d_in[7];
  const float* atts    = (const float*)d_in[8];
  const float* attd    = (const float*)d_in[9];
  const float* bgat    = (const float*)d_in[10];
  const float* Wgcn    = (const float*)d_in[11];
  const float* bgcn    = (const float*)d_in[12];
  const float* Wt      = (const float*)d_in[13];
  const float* bt      = (const float*)d_in[14];
  const float* W1      = (const float*)d_in[15];
  const float* b1      = (const float*)d_in[16];
  const float* W2      = (const float*)d_in[17];
  const float* b2      = (const float*)d_in[18];
  float* out = (float*)d_out;

  float* w = (float*)d_ws;
  size_t o = 0;
  auto alloc = [&](size_t n) { float* p = w + o; o += (n + 63) & ~(size_t)63; return p; };
  float* X    = alloc((size_t)PN * DN);
  float* Xp   = alloc((size_t)PN * HIDN);
  float* a_s  = alloc((size_t)PN * HN);
  float* a_d  = alloc((size_t)PN * HN);
  float* mbuf = alloc((size_t)PN * HN);
  float* denb = alloc((size_t)PN * HN);
  float* exb  = alloc((size_t)ETOT * HN);
  float* gacc = alloc((size_t)PN * HIDN);
  float* fin  = alloc((size_t)PN * FINN);
  float* vbuf = alloc((size_t)PN * DN);
  float* ctx  = alloc((size_t)ZN * DN);
  float* XW   = alloc((size_t)ZN * DN);
  float* dinv = alloc((size_t)ZN);
  float* embp = alloc((size_t)ZPAD * DN);
  float* esq  = alloc((size_t)ZPAD);
  int*   zi   = (int*)alloc((size_t)PN);
  unsigned* adjm = (unsigned*)alloc((size_t)ZN * ADJW);

  const int B = 256;
  auto G = [](int n, int b) { return (n + b - 1) / b; };

  // -------- precompute (time-independent)
  pre_zero_kernel<<<G(ZN * DN, B), B, 0, stream>>>(dinv, ctx, adjm);
  pre_deg_kernel<<<G(ELOC, B), B, 0, stream>>>(le, dinv);
  pre_dinv_kernel<<<G(ZN, B), B, 0, stream>>>(dinv);
  pre_xw_kernel<<<G(ZN * DN, B), B, 0, stream>>>(emb, Wgcn, XW);
  pre_gcn_scatter_kernel<<<G((ELOC + ZN) * DN, B), B, 0, stream>>>(le, XW, dinv, ctx);
  pre_adj_kernel<<<G(ELOC + ZN, B), B, 0, stream>>>(le, adjm);
  pre_embpad_kernel<<<G(ZPAD * DN, B), B, 0, stream>>>(emb, embp);
  pre_esq_kernel<<<G(ZPAD, B), B, 0, stream>>>(emb, esq);
  pre_basep_x0_kernel<<<G(PN * DN, B), B, 0, stream>>>(x_zones, emb, Wpe, bpe, fin, X, out);

  // -------- Euler steps
  for (int ti = 0; ti < 9; ++ti) {
    xp_kernel<<<PN / 16, B, 0, stream>>>(X, Wgat, Xp);
    attn_kernel<<<G(PN * HN, B), B, 0, stream>>>(Xp, atts, attd, a_s, a_d);
    fill_kernel<<<G(PN * HIDN, B), B, 0, stream>>>(mbuf, denb, gacc);
    edge_max_kernel<<<G(ETOT * HN, B), B, 0, stream>>>(epp, a_s, a_d, mbuf);
    edge_exp_kernel<<<G(ETOT * HN, B), B, 0, stream>>>(epp, a_s, a_d, mbuf, exb, denb);
    edge_scatter_kernel<<<G(ETOT * DN, B), B, 0, stream>>>(epp, exb, denb, Xp, gacc);
    argmin_kernel<<<G(PN / 16, 8), B, 0, stream>>>(X, embp, esq, zi);
    assemble_kernel<<<G(PN * DN, B), B, 0, stream>>>(gacc, bgat, ctx, bgcn, zi, Wt, bt,
                                                     times, ti, fin);
    mlp_kernel<<<G(PN / 16, 8), B, 0, stream>>>(fin, W1, b1, W2, b2, vbuf);
    flow_kernel<<<G(PN, 8), B, 0, stream>>>(X, vbuf, zi, emb, adjm, times, ti, out);
  }
}